// IvySPT_16698832846925
// MI455X (gfx1250) — compile-verified
//
#include <hip/hip_runtime.h>
#include <hip/hip_bf16.h>

typedef __attribute__((ext_vector_type(2))) float v2f;
typedef __attribute__((ext_vector_type(8))) float v8f;

#define B_ 8
#define N_ 1024
#define Q_ 128
#define K_ 16
#define M_ (N_ + Q_)

// Giles (2010) single-precision erfinv approximation.
__device__ __forceinline__ float erfinv_f(float x) {
    float w = -__logf((1.0f - x) * (1.0f + x));
    float p;
    if (w < 5.0f) {
        w -= 2.5f;
        p = 2.81022636e-08f;
        p = 3.43273939e-07f + p * w;
        p = -3.5233877e-06f + p * w;
        p = -4.39150654e-06f + p * w;
        p = 0.00021858087f + p * w;
        p = -0.00125372503f + p * w;
        p = -0.00417768164f + p * w;
        p = 0.246640727f + p * w;
        p = 1.50140941f + p * w;
    } else {
        w = sqrtf(w) - 3.0f;
        p = -0.000200214257f;
        p = 0.000100950558f + p * w;
        p = 0.00134934322f + p * w;
        p = -0.00367342844f + p * w;
        p = 0.00573950773f + p * w;
        p = -0.0076224613f + p * w;
        p = 0.00943887047f + p * w;
        p = 1.00167406f + p * w;
        p = 2.83297682f + p * w;
    }
    return p * x;
}

// Fold bandwidth + elliptical scales + log2(e) into per-kernel exponent
// coefficients so the inner loop can use exp2 (raw v_exp_f32, no pre-mul):
// coefs[k]      = -log2(e) * sx_k / (2*bw_k^2*tr_k)   (multiplies dx^2)
// coefs[16 + k] = -log2(e) * sy_k / (2*bw_k^2*tr_k)   (multiplies dy^2)
__global__ void ivyspt_prep(const float* __restrict__ ls, float* __restrict__ coefs) {
    int k = threadIdx.x;
    if (k < K_) {
        const float log2e = 1.44269504088896340736f;
        float bw = erfinv_f((float)(k + 1) / (float)(K_ + 1)) * 1.41421356237f;
        float sx = __expf(ls[2 * k + 0]);
        float sy = __expf(ls[2 * k + 1]);
        float tr = sx + sy;
        float c  = log2e / (2.0f * bw * bw * tr);
        coefs[k]      = -sx * c;
        coefs[K_ + k] = -sy * c;
    }
}

__global__ __launch_bounds__(256) void ivyspt_main(
    const float* __restrict__ lm_s,  const float* __restrict__ ttm_s,
    const float* __restrict__ tv_s,  const float* __restrict__ lm_q,
    const float* __restrict__ ttm_q, const float* __restrict__ coefs,
    const float* __restrict__ ln_s_w, const float* __restrict__ ln_s_b,
    const float* __restrict__ ln_q_w, const float* __restrict__ ln_q_b,
    const float* __restrict__ pe_log_scale, float* __restrict__ out)
{
    __shared__ __align__(16) float s_x[N_];
    __shared__ __align__(16) float s_y[N_];
    __shared__ __align__(16) float s_tv[N_];

    const int b   = blockIdx.x;
    const int tid = threadIdx.x;

    // Stage surface points into LDS (reused by all 8 waves x 64 n-chunks).
    for (int i = tid; i < N_; i += 256) {
        s_x[i]  = lm_s [b * N_ + i];
        s_y[i]  = ttm_s[b * N_ + i];
        s_tv[i] = tv_s [b * N_ + i];
    }
    __syncthreads();

    const int wave = tid >> 5;      // wave32: 8 waves / 256-thread block
    const int l    = tid & 31;
    const int kk   = l & 15;        // kernel index (WMMA column)
    const int lh   = l >> 4;        // lane half
    const int m    = blockIdx.y * 8 + wave;   // 0..1151, wave-uniform

    float xm, ym;
    if (m < N_) { xm = lm_s[b * N_ + m];        ym = ttm_s[b * N_ + m]; }
    else        { xm = lm_q[b * Q_ + (m - N_)]; ym = ttm_q[b * Q_ + (m - N_)]; }

    // B matrix (4x16, K x kernels): rows 0/1 = coefficients, rows 2/3 = 0.
    v2f Bm;
    Bm.x = lh ? 0.0f : coefs[kk];
    Bm.y = lh ? 0.0f : coefs[K_ + kk];

    float wsum = 0.0f, norm = 0.0f;

    for (int n0 = 0; n0 < N_; n0 += 16) {
        // A matrix (16x4, n-rows x channels): lanes 0-15 rows, slots 2/3 zero.
        float dx = xm - s_x[n0 + kk];
        float dy = ym - s_y[n0 + kk];
        v2f A;
        A.x = lh ? 0.0f : dx * dx;
        A.y = lh ? 0.0f : dy * dy;

        v8f c = {};
        // D[n_row, k_col] = -log2(e)*(a_k*dx^2 + b_k*dy^2): reference einsum tile.
        c = __builtin_amdgcn_wmma_f32_16x16x4_f32(
                /*neg_a=*/false, A, /*neg_b=*/false, Bm,
                /*c_mod=*/(short)0, c, /*reuse_a=*/false, /*reuse_b=*/false);

        // tv weights for this lane-half's 8 rows (n0 + lh*8 .. +7).
        const float4* tvp = (const float4*)&s_tv[n0 + lh * 8];
        float4 t0 = tvp[0];
        float4 t1 = tvp[1];

        // log2(e) folded into coefs -> raw native v_exp_f32, no pre-mul.
        float e0 = __builtin_amdgcn_exp2f(c[0]);
        float e1 = __builtin_amdgcn_exp2f(c[1]);
        float e2 = __builtin_amdgcn_exp2f(c[2]);
        float e3 = __builtin_amdgcn_exp2f(c[3]);
        float e4 = __builtin_amdgcn_exp2f(c[4]);
        float e5 = __builtin_amdgcn_exp2f(c[5]);
        float e6 = __builtin_amdgcn_exp2f(c[6]);
        float e7 = __builtin_amdgcn_exp2f(c[7]);

        wsum += e0 * t0.x + e1 * t0.y + e2 * t0.z + e3 * t0.w
              + e4 * t1.x + e5 * t1.y + e6 * t1.z + e7 * t1.w;
        norm += (e0 + e1 + e2 + e3) + (e4 + e5 + e6 + e7);
    }

    // Fold the two lane-halves (each covered 8 of 16 rows per chunk).
    wsum += __shfl_xor(wsum, 16, 32);
    norm += __shfl_xor(norm, 16, 32);
    float emb = wsum / norm;

    // LayerNorm across K=16 (lanes kk=0..15; both halves compute identically).
    float s = emb;
    s += __shfl_xor(s, 1, 32);
    s += __shfl_xor(s, 2, 32);
    s += __shfl_xor(s, 4, 32);
    s += __shfl_xor(s, 8, 32);
    float mean = s * (1.0f / 16.0f);
    float d0 = emb - mean;
    float v = d0 * d0;
    v += __shfl_xor(v, 1, 32);
    v += __shfl_xor(v, 2, 32);
    v += __shfl_xor(v, 4, 32);
    v += __shfl_xor(v, 8, 32);
    float rstd = rsqrtf(v * (1.0f / 16.0f) + 1e-5f);

    float lw = (m < N_) ? ln_s_w[kk] : ln_q_w[kk];
    float lb = (m < N_) ? ln_s_b[kk] : ln_q_b[kk];
    float lnv = d0 * rstd * lw + lb;

    // Positional embedding: kk = 4*i + j, j in {sin x, cos x, sin y, cos y}.
    int   i4    = kk >> 2;
    int   j     = kk & 3;
    float div   = __expf(pe_log_scale[0] * (float)i4 * 0.25f);
    float coord = (j < 2) ? xm : ym;
    float arg   = coord / div;
    float pe    = (j & 1) ? __cosf(arg) : __sinf(arg);
    float res   = lnv + 1.41421356237f * pe;

    if (l < 16)
        out[(b * M_ + m) * K_ + kk] = res;
}

extern "C" void kernel_launch(void* const* d_in, const int* in_sizes, int n_in,
                              void* d_out, int out_size, void* d_ws, size_t ws_size,
                              hipStream_t stream) {
    const float* lm_s   = (const float*)d_in[0];
    const float* ttm_s  = (const float*)d_in[1];
    const float* tv_s   = (const float*)d_in[2];
    const float* lm_q   = (const float*)d_in[3];
    const float* ttm_q  = (const float*)d_in[4];
    const float* lsk    = (const float*)d_in[5];
    const float* ln_s_w = (const float*)d_in[6];
    const float* ln_s_b = (const float*)d_in[7];
    const float* ln_q_w = (const float*)d_in[8];
    const float* ln_q_b = (const float*)d_in[9];
    const float* pe_ls  = (const float*)d_in[10];
    float* out   = (float*)d_out;
    float* coefs = (float*)d_ws;   // 2*K floats

    ivyspt_prep<<<1, 32, 0, stream>>>(lsk, coefs);

    dim3 grid(B_, M_ / 8);
    ivyspt_main<<<grid, 256, 0, stream>>>(lm_s, ttm_s, tv_s, lm_q, ttm_q, coefs,
                                          ln_s_w, ln_s_b, ln_q_w, ln_q_b,
                                          pe_ls, out);
}